// VariationalLinearEncoder_89335319757133
// MI455X (gfx1250) — compile-verified
//
#include <hip/hip_runtime.h>

typedef __attribute__((ext_vector_type(2))) float v2f;
typedef __attribute__((ext_vector_type(8))) float v8f;

#define D_IN  48
#define D_OUT 16

// ---------------- degree kernels ----------------

__global__ void k_init_deg(float* __restrict__ deg, int n) {
    int i = blockIdx.x * blockDim.x + threadIdx.x;
    if (i < n) deg[i] = 1.0f;   // self-loop contributes 1
}

__global__ void k_accum_deg(const int* __restrict__ dst, float* __restrict__ deg, int e) {
    int i = blockIdx.x * blockDim.x + threadIdx.x;
    if (i < e) atomicAdd(&deg[dst[i]], 1.0f);
}

__global__ void k_deg_to_dinv(float* __restrict__ deg, int n) {
    int i = blockIdx.x * blockDim.x + threadIdx.x;
    if (i < n) deg[i] = rsqrtf(deg[i]);   // in place: deg -> deg^{-1/2}
}

// ---------------- fused projection (WMMA f32 16x16x4) ----------------
// Computes y = x @ [W_mu | W_logstd]  (N x 32), stores y to workspace, and
// initializes d_out with the self-loop term + bias:
//   out_mu[i][c] = y_mu[i][c] * dinv[i]^2 + b_mu[c]   (same for logstd)
// One wave handles a 16-node tile: 12 K-steps x 2 WMMAs.

__global__ void __launch_bounds__(256)
k_gemm_wmma(const float* __restrict__ x,
            const float* __restrict__ Wmu, const float* __restrict__ bmu,
            const float* __restrict__ Wls, const float* __restrict__ bls,
            const float* __restrict__ dinv,
            float* __restrict__ y,          // [N,32]
            float* __restrict__ out,        // mu [N,16] then logstd [N,16]
            int n)
{
    const int lane = threadIdx.x & 31;
    const int wave = (blockIdx.x * blockDim.x + threadIdx.x) >> 5;
    const int node_base = wave * 16;
    if (node_base >= n) return;             // wave-uniform: EXEC all-ones for WMMA

    const int half = lane >> 4;             // 0: lanes 0-15, 1: lanes 16-31
    const int l16  = lane & 15;

    v8f cmu = {};
    v8f cls = {};

    // A fragment: M = l16, VGPR0 holds K = kbase + 2*half, VGPR1 holds K+1.
    const float* xrow = x + (size_t)(node_base + l16) * D_IN;

    #pragma unroll
    for (int ks = 0; ks < D_IN; ks += 4) {
        const int kA = ks + half * 2;
        v2f a;  a.x  = xrow[kA];              a.y  = xrow[kA + 1];
        v2f bm; bm.x = Wmu[kA * D_OUT + l16]; bm.y = Wmu[(kA + 1) * D_OUT + l16];
        v2f bl; bl.x = Wls[kA * D_OUT + l16]; bl.y = Wls[(kA + 1) * D_OUT + l16];

        cmu = __builtin_amdgcn_wmma_f32_16x16x4_f32(
                  false, a, false, bm, (short)0, cmu, false, false);
        cls = __builtin_amdgcn_wmma_f32_16x16x4_f32(
                  false, a, false, bl, (short)0, cls, false, false);
    }

    const float bmu_c = bmu[l16];
    const float bls_c = bls[l16];
    const size_t ls_off = (size_t)n * D_OUT;

    // C/D layout: VGPR r holds (M = r + 8*half, N = l16)
    #pragma unroll
    for (int r = 0; r < 8; ++r) {
        const int node = node_base + r + half * 8;
        const float di = dinv[node];
        const float d2 = di * di;            // deg^{-1}
        const float vmu = cmu[r];
        const float vls = cls[r];
        y[(size_t)node * 32 + l16]      = vmu;
        y[(size_t)node * 32 + 16 + l16] = vls;
        out[(size_t)node * D_OUT + l16]          = vmu * d2 + bmu_c;
        out[ls_off + (size_t)node * D_OUT + l16] = vls * d2 + bls_c;
    }
}

// ---------------- edge scatter ----------------
// 8 threads per edge; thread cg handles 4 of the 32 projected columns.
// out += dinv[src]*dinv[dst] * y[src]   accumulated at dst via f32 atomics.

__global__ void __launch_bounds__(256)
k_scatter(const int* __restrict__ src, const int* __restrict__ dst,
          const float* __restrict__ dinv, const float* __restrict__ y,
          float* __restrict__ out, int e, int n)
{
    long long gid = (long long)blockIdx.x * blockDim.x + threadIdx.x;
    if (gid >= (long long)e * 8) return;
    const int ei = (int)(gid >> 3);
    const int cg = (int)(gid & 7);

    const int s = src[ei];
    const int d = dst[ei];
    const float norm = dinv[s] * dinv[d];

    const float4 v = *(const float4*)(y + (size_t)s * 32 + cg * 4);

    float* base = (cg < 4)
        ? (out + (size_t)d * D_OUT + cg * 4)
        : (out + (size_t)n * D_OUT + (size_t)d * D_OUT + (cg - 4) * 4);

    atomicAdd(base + 0, v.x * norm);
    atomicAdd(base + 1, v.y * norm);
    atomicAdd(base + 2, v.z * norm);
    atomicAdd(base + 3, v.w * norm);
}

// ---------------- launch ----------------

extern "C" void kernel_launch(void* const* d_in, const int* in_sizes, int n_in,
                              void* d_out, int out_size, void* d_ws, size_t ws_size,
                              hipStream_t stream) {
    const float* x   = (const float*)d_in[0];
    const int*   ei  = (const int*)  d_in[1];
    const float* Wmu = (const float*)d_in[2];
    const float* bmu = (const float*)d_in[3];
    const float* Wls = (const float*)d_in[4];
    const float* bls = (const float*)d_in[5];
    float* out = (float*)d_out;

    const int n = in_sizes[0] / D_IN;       // 100000
    const int e = in_sizes[1] / 2;          // 1600000

    const int* src = ei;
    const int* dst = ei + e;

    float* deg = (float*)d_ws;              // N floats; becomes dinv in place
    float* y   = deg + n;                   // N*32 floats

    k_init_deg  <<<(n + 255) / 256, 256, 0, stream>>>(deg, n);
    k_accum_deg <<<(e + 255) / 256, 256, 0, stream>>>(dst, deg, e);
    k_deg_to_dinv<<<(n + 255) / 256, 256, 0, stream>>>(deg, n);

    const int waves  = (n + 15) / 16;       // 6250 node tiles
    const int blocks = (waves + 7) / 8;     // 8 waves / 256-thread block
    k_gemm_wmma<<<blocks, 256, 0, stream>>>(x, Wmu, bmu, Wls, bls, deg, y, out, n);

    const long long tot = (long long)e * 8;
    k_scatter<<<(int)((tot + 255) / 256), 256, 0, stream>>>(src, dst, deg, y, out, e, n);
}